// Selector_46093589021392
// MI455X (gfx1250) — compile-verified
//
#include <hip/hip_runtime.h>
#include <hip/hip_bf16.h>
#include <math.h>

// ---------------- problem constants (from reference) ----------------
#define N_IMG   16
#define N_CLS   80
#define HH      160
#define WW      160
#define HW      (HH * WW)          // 25600
#define POST_N  100                // fpn_post_nms_top_n
#define PRE_THR 0.01f

// ---------------- algorithm constants ----------------
#define NBINS   4096               // score in (0,1) -> bin = floor(s*NBINS)
#define HWB     1280               // hw positions per block (HW/HWB = 20)
#define CSPLIT  2                  // class-dimension split per image
#define CPB     (N_CLS / CSPLIT)   // 40 classes per block
#define TB      256                // threads per block (8 waves of 32)
#define CAP     8192               // candidate capacity per image (pow2)
#define CHUNK   (NBINS / TB)       // 16 bins per thread in threshold scan

// ---------------- gfx1250 async global->LDS staging ----------------
#if defined(__has_builtin)
#if __has_builtin(__builtin_amdgcn_global_load_async_to_lds_b128) && \
    __has_builtin(__builtin_amdgcn_s_wait_asynccnt)
#define USE_ASYNC_LDS 1
#endif
#endif

typedef int v4i __attribute__((ext_vector_type(4)));
typedef __attribute__((address_space(1))) v4i GV4;   // global int4
typedef __attribute__((address_space(3))) v4i LV4;   // LDS int4

// Stage `count` floats (count % 4 == 0, 16B-aligned src) from global into LDS.
__device__ __forceinline__ void stage_to_lds(const float* __restrict__ gsrc,
                                             float* lds, int count) {
#ifdef USE_ASYNC_LDS
    for (int v = threadIdx.x; v < count / 4; v += blockDim.x) {
        __builtin_amdgcn_global_load_async_to_lds_b128(
            (GV4*)(gsrc + 4 * v), (LV4*)(lds + 4 * v), 0, 0);
    }
    __builtin_amdgcn_s_wait_asynccnt(0);
#else
    for (int v = threadIdx.x; v < count / 4; v += blockDim.x) {
        float4 t = ((const float4*)gsrc)[v];
        ((float4*)lds)[v] = t;
    }
#endif
}

// Fast sigmoid: prefer gfx1250 v_tanh_f32 (3 VALU ops incl. 1 trans).
#if defined(__has_builtin)
#if __has_builtin(__builtin_amdgcn_tanhf)
#define HAVE_TANH 1
#endif
#endif

__device__ __forceinline__ float sigmoidf(float x) {
#ifdef HAVE_TANH
    return fmaf(0.5f, __builtin_amdgcn_tanhf(0.5f * x), 0.5f);
#else
    return __builtin_amdgcn_rcpf(1.0f + __expf(-x));
#endif
}

__device__ __forceinline__ int score_bin(float s) {
    int b = (int)(s * (float)NBINS);
    return b > (NBINS - 1) ? (NBINS - 1) : (b < 0 ? 0 : b);
}

// ------------------------------------------------------------------
// K1: per-image score histogram. grid = (HW/HWB, CSPLIT, N_IMG)
// ------------------------------------------------------------------
__global__ void __launch_bounds__(TB)
hist_kernel(const float* __restrict__ cls, const float* __restrict__ ctr,
            unsigned int* __restrict__ hist) {
    __shared__ unsigned int lh[NBINS];           // 16 KB
    __shared__ __align__(16) float sctr[HWB];    // 5 KB

    const int n   = blockIdx.z;
    const int c0  = blockIdx.y * CPB;
    const int hw0 = blockIdx.x * HWB;

    for (int i = threadIdx.x; i < NBINS; i += TB) lh[i] = 0u;

    // stage raw centerness into LDS (async gfx1250 path), sigmoid in place
    stage_to_lds(ctr + (size_t)n * HW + hw0, sctr, HWB);
    __syncthreads();
    for (int i = threadIdx.x; i < HWB; i += TB) sctr[i] = sigmoidf(sctr[i]);
    __syncthreads();

    const float* clsb = cls + ((size_t)n * N_CLS + c0) * HW + hw0;
    for (int c = 0; c < CPB; ++c) {
        const float* p = clsb + (size_t)c * HW;
        if (c + 1 < CPB) __builtin_prefetch(p + HW, 0, 0);  // global_prefetch_b8
        for (int v = threadIdx.x; v < HWB / 4; v += TB) {
            float4 x4 = ((const float4*)p)[v];
            float4 t4 = ((const float4*)sctr)[v];
            float sc0 = sigmoidf(x4.x), sc1 = sigmoidf(x4.y);
            float sc2 = sigmoidf(x4.z), sc3 = sigmoidf(x4.w);
            if (sc0 > PRE_THR) atomicAdd(&lh[score_bin(sc0 * t4.x)], 1u);
            if (sc1 > PRE_THR) atomicAdd(&lh[score_bin(sc1 * t4.y)], 1u);
            if (sc2 > PRE_THR) atomicAdd(&lh[score_bin(sc2 * t4.z)], 1u);
            if (sc3 > PRE_THR) atomicAdd(&lh[score_bin(sc3 * t4.w)], 1u);
        }
    }
    __syncthreads();

    unsigned int* gh = hist + (size_t)n * NBINS;
    for (int i = threadIdx.x; i < NBINS; i += TB)
        if (lh[i]) atomicAdd(&gh[i], lh[i]);
}

// ------------------------------------------------------------------
// K2: per-image threshold bin (parallel top-down suffix scan)
// grid = N_IMG blocks of TB threads; thread t owns CHUNK bins from the top
// ------------------------------------------------------------------
__global__ void __launch_bounds__(TB)
thresh_kernel(const unsigned int* __restrict__ hist, int* __restrict__ tbin) {
    __shared__ unsigned int csum[TB];
    __shared__ unsigned int pre[TB];
    __shared__ int result;

    const int n = blockIdx.x;
    const int t = threadIdx.x;
    const unsigned int* h = hist + (size_t)n * NBINS;

    // chunk t covers bins [NBINS - CHUNK*(t+1), NBINS-1 - CHUNK*t] (t=0 = top)
    const int lo = NBINS - CHUNK * (t + 1);
    unsigned int s = 0;
    for (int j = 0; j < CHUNK; ++j) s += h[lo + j];
    csum[t] = s;
    pre[t]  = s;
    if (t == 0) result = 0;
    __syncthreads();

    for (int off = 1; off < TB; off <<= 1) {   // Hillis-Steele inclusive scan
        unsigned int v = (t >= off) ? pre[t - off] : 0u;
        __syncthreads();
        pre[t] += v;
        __syncthreads();
    }

    const unsigned int incl = pre[t];
    const unsigned int excl = incl - csum[t];
    if (excl < (unsigned)POST_N && incl >= (unsigned)POST_N) {
        unsigned int cum = excl;
        const int btop = NBINS - 1 - CHUNK * t;
        for (int j = 0; j < CHUNK; ++j) {
            cum += h[btop - j];
            if (cum >= (unsigned)POST_N) { result = btop - j; break; }
        }
    }
    __syncthreads();
    if (t == 0) tbin[n] = result;   // 0 if fewer than POST_N candidates total
}

// ------------------------------------------------------------------
// K3: collect (score, flat index) for all entries in bins >= threshold
// ------------------------------------------------------------------
__global__ void __launch_bounds__(TB)
collect_kernel(const float* __restrict__ cls, const float* __restrict__ ctr,
               const int* __restrict__ tbin, unsigned int* __restrict__ ccount,
               float* __restrict__ cscore, unsigned int* __restrict__ cidx) {
    __shared__ __align__(16) float sctr[HWB];

    const int n   = blockIdx.z;
    const int c0  = blockIdx.y * CPB;
    const int hw0 = blockIdx.x * HWB;
    const int tb  = tbin[n];

    stage_to_lds(ctr + (size_t)n * HW + hw0, sctr, HWB);
    __syncthreads();
    for (int i = threadIdx.x; i < HWB; i += TB) sctr[i] = sigmoidf(sctr[i]);
    __syncthreads();

    const float* clsb = cls + ((size_t)n * N_CLS + c0) * HW + hw0;
    for (int c = 0; c < CPB; ++c) {
        const float* p = clsb + (size_t)c * HW;
        if (c + 1 < CPB) __builtin_prefetch(p + HW, 0, 0);
        for (int v = threadIdx.x; v < HWB / 4; v += TB) {
            float4 x4 = ((const float4*)p)[v];
            float4 t4 = ((const float4*)sctr)[v];
            #pragma unroll
            for (int e = 0; e < 4; ++e) {
                float x = (e == 0) ? x4.x : (e == 1) ? x4.y : (e == 2) ? x4.z : x4.w;
                float t = (e == 0) ? t4.x : (e == 1) ? t4.y : (e == 2) ? t4.z : t4.w;
                float sc = sigmoidf(x);
                if (sc > PRE_THR) {
                    float s = sc * t;
                    if (score_bin(s) >= tb) {
                        unsigned int pos = atomicAdd(&ccount[n], 1u);
                        if (pos < CAP) {
                            cscore[(size_t)n * CAP + pos] = s;
                            cidx[(size_t)n * CAP + pos] =
                                (unsigned)((hw0 + 4 * v + e) * N_CLS + (c0 + c));
                        }
                    }
                }
            }
        }
    }
}

// ------------------------------------------------------------------
// K4: exact top-100 via LDS bitonic sort on packed keys + box decode.
// Key = (score_bits << 32) | ~idx : descending sort == score desc, idx asc
// (matches top_k tie-break). grid = N_IMG blocks of TB threads.
// ------------------------------------------------------------------
__global__ void __launch_bounds__(TB)
topk_decode_kernel(const float* __restrict__ cscore, const unsigned int* __restrict__ cidx,
                   const unsigned int* __restrict__ ccount,
                   const float* __restrict__ loc, const float* __restrict__ reg,
                   const int* __restrict__ imsz, float* __restrict__ out) {
    __shared__ unsigned long long key[CAP];   // 64 KB

    const int n   = blockIdx.x;
    const int tid = threadIdx.x;
    unsigned int cnt = ccount[n];
    const int m = (int)(cnt < (unsigned)CAP ? cnt : (unsigned)CAP);

    int M = 128;
    while (M < m) M <<= 1;                     // next pow2 >= m (<= CAP)

    for (int i = tid; i < M; i += TB) {
        if (i < m) {
            unsigned int sb = __float_as_uint(cscore[(size_t)n * CAP + i]);
            unsigned int id = cidx[(size_t)n * CAP + i];
            key[i] = ((unsigned long long)sb << 32) | (unsigned int)(~id);
        } else {
            key[i] = 0ull;                     // pads sort last (scores > 0)
        }
    }
    __syncthreads();

    for (int k = 2; k <= M; k <<= 1) {
        for (int j = k >> 1; j > 0; j >>= 1) {
            for (int i = tid; i < M; i += TB) {
                int l = i ^ j;
                if (l > i) {
                    unsigned long long a = key[i], b = key[l];
                    bool desc = ((i & k) == 0);
                    if (desc ? (a < b) : (a > b)) { key[i] = b; key[l] = a; }
                }
            }
            __syncthreads();
        }
    }

    const int meff = m < POST_N ? m : POST_N;
    const float wl = (float)(imsz[n * 2 + 1] - 1);
    const float hl = (float)(imsz[n * 2 + 0] - 1);
    for (int k = tid; k < POST_N; k += TB) {
        float o0 = 0.f, o1 = 0.f, o2 = 0.f, o3 = 0.f, o4 = 0.f, o5 = 0.f;
        if (k < meff) {
            unsigned long long kk = key[k];
            float sc = __uint_as_float((unsigned int)(kk >> 32));
            unsigned int id = ~(unsigned int)(kk & 0xFFFFFFFFull);
            unsigned int hw = id / N_CLS;
            unsigned int c  = id - hw * N_CLS;
            float lx = loc[(size_t)hw * 2 + 0];
            float ly = loc[(size_t)hw * 2 + 1];
            const float* r = reg + (size_t)n * 4 * HW + hw;
            float r0 = r[0], r1 = r[HW], r2 = r[2 * HW], r3 = r[3 * HW];
            o0 = fminf(fmaxf(lx - r0, 0.f), wl);
            o1 = fminf(fmaxf(ly - r1, 0.f), hl);
            o2 = fminf(fmaxf(lx + r2, 0.f), wl);
            o3 = fminf(fmaxf(ly + r3, 0.f), hl);
            o4 = sc;
            o5 = (float)(c + 1);
        }
        float* op = out + ((size_t)n * POST_N + k) * 6;
        op[0] = o0; op[1] = o1; op[2] = o2; op[3] = o3; op[4] = o4; op[5] = o5;
    }
}

// ------------------------------------------------------------------
extern "C" void kernel_launch(void* const* d_in, const int* in_sizes, int n_in,
                              void* d_out, int out_size, void* d_ws, size_t ws_size,
                              hipStream_t stream) {
    (void)in_sizes; (void)n_in; (void)out_size; (void)ws_size;
    const float* locations = (const float*)d_in[0];   // (HW, 2)
    const float* box_cls   = (const float*)d_in[1];   // (N, C, H, W)
    const float* box_reg   = (const float*)d_in[2];   // (N, 4, H, W)
    const float* ctrness   = (const float*)d_in[3];   // (N, 1, H, W)
    const int*   imsz      = (const int*)d_in[4];     // (N, 2)
    float*       out       = (float*)d_out;           // (N, 100, 6)

    // workspace layout
    unsigned char* ws = (unsigned char*)d_ws;
    unsigned int* hist  = (unsigned int*)ws;                                  // N*NBINS u32 (256 KB)
    int*          tbin  = (int*)(ws + (size_t)N_IMG * NBINS * 4);             // N i32
    unsigned int* ccnt  = (unsigned int*)(ws + (size_t)N_IMG * NBINS * 4 + 64);
    float*        cscr  = (float*)(ws + (size_t)N_IMG * NBINS * 4 + 128);     // N*CAP f32
    unsigned int* cid   = (unsigned int*)((unsigned char*)cscr + (size_t)N_IMG * CAP * 4);

    // zero histogram + counters every call (graph-capturable)
    (void)hipMemsetAsync(ws, 0, (size_t)N_IMG * NBINS * 4 + 128, stream);

    dim3 gScan(HW / HWB, CSPLIT, N_IMG);   // 20 x 2 x 16 = 640 blocks
    hist_kernel<<<gScan, TB, 0, stream>>>(box_cls, ctrness, hist);
    thresh_kernel<<<N_IMG, TB, 0, stream>>>(hist, tbin);
    collect_kernel<<<gScan, TB, 0, stream>>>(box_cls, ctrness, tbin, ccnt, cscr, cid);
    topk_decode_kernel<<<N_IMG, TB, 0, stream>>>(cscr, cid, ccnt, locations, box_reg,
                                                 imsz, out);
}